// ChipCircuit_2929167696517
// MI455X (gfx1250) — compile-verified
//
#include <hip/hip_runtime.h>
#include <hip/hip_bf16.h>

typedef __attribute__((ext_vector_type(2))) float v2f;
typedef __attribute__((ext_vector_type(8))) float v8f;

// D = A(16x4,f32) * B(4x16,f32) + C ; A/B neg must be 0 for F32 per ISA NEG table.
#define WMMA4(a, b, c) \
  __builtin_amdgcn_wmma_f32_16x16x4_f32(false, (a), false, (b), (short)0, (c), false, false)

// ---------------------------------------------------------------------------
// Generic f32 WMMA GEMM: C(MxN) = A(MxK) @ B(KxN) + bias, optional SiLU.
// Block = 128 threads (4 waves), 64x64 output tile, K staged in LDS by 16.
// ---------------------------------------------------------------------------
#define GS_A 17
#define GS_B 72

__launch_bounds__(128)
__global__ void gemm_f32_kernel(const float* __restrict__ A, const float* __restrict__ B,
                                const float* __restrict__ bias, float* __restrict__ C,
                                int M, int N, int K, int do_silu) {
  __shared__ float As[64 * GS_A];  // [m][k]
  __shared__ float Bs[16 * GS_B];  // [k][n]
  const int tid  = threadIdx.x;
  const int lane = tid & 31;
  const int wave = tid >> 5;
  const int bm = blockIdx.x * 64;
  const int bn = blockIdx.y * 64;
  const int mlo = lane & 15;
  const int kb  = (lane < 16) ? 0 : 2;

  v8f acc[4];
  const v8f zero8 = {0.f, 0.f, 0.f, 0.f, 0.f, 0.f, 0.f, 0.f};
#pragma unroll
  for (int nt = 0; nt < 4; ++nt) acc[nt] = zero8;

  for (int k0 = 0; k0 < K; k0 += 16) {
    for (int e = tid; e < 64 * 16; e += 128) {
      int ml = e >> 4, kl = e & 15;
      int gm = bm + ml, gk = k0 + kl;
      As[ml * GS_A + kl] = (gm < M && gk < K) ? A[gm * K + gk] : 0.0f;
    }
    for (int e = tid; e < 16 * 64; e += 128) {
      int kl = e >> 6, nl = e & 63;
      int gk = k0 + kl, gn = bn + nl;
      Bs[kl * GS_B + nl] = (gk < K && gn < N) ? B[gk * N + gn] : 0.0f;
    }
    __syncthreads();
#pragma unroll
    for (int kk = 0; kk < 4; ++kk) {
      const int k4 = kk * 4 + kb;
      v2f a;
      a.x = As[(wave * 16 + mlo) * GS_A + k4];
      a.y = As[(wave * 16 + mlo) * GS_A + k4 + 1];
#pragma unroll
      for (int nt = 0; nt < 4; ++nt) {
        v2f bf;
        bf.x = Bs[k4 * GS_B + nt * 16 + mlo];
        bf.y = Bs[(k4 + 1) * GS_B + nt * 16 + mlo];
        acc[nt] = WMMA4(a, bf, acc[nt]);
      }
    }
    __syncthreads();
  }
#pragma unroll
  for (int nt = 0; nt < 4; ++nt) {
    int gn = bn + nt * 16 + mlo;
    if (gn >= N) continue;
    float bv = bias ? bias[gn] : 0.0f;
#pragma unroll
    for (int v = 0; v < 8; ++v) {
      int gm = bm + wave * 16 + v + ((lane < 16) ? 0 : 8);
      if (gm < M) {
        float val = acc[nt][v] + bv;
        if (do_silu) val = val / (1.0f + __expf(-val));
        C[gm * N + gn] = val;
      }
    }
  }
}

// ---------------------------------------------------------------------------
// Fused: build H_eff from theta via sparse Pauli structure (no pauli tensor
// read), then psi = exp(i*H) e0 by scaling-and-squaring with f32 WMMA.
// One block (4 waves) per batch element. All matrices live in LDS.
// ---------------------------------------------------------------------------
#define ES 68  // LDS row stride (floats) for 64x64 matrices (bank padding)

__device__ __forceinline__ void cmm_accum(const float* Xr, const float* Xi,
                                          const float* Yr, const float* Yi,
                                          int mrow, int kb, int mlo,
                                          v8f aR[4], v8f aI[4]) {
  const v8f zero8 = {0.f, 0.f, 0.f, 0.f, 0.f, 0.f, 0.f, 0.f};
#pragma unroll
  for (int nt = 0; nt < 4; ++nt) { aR[nt] = zero8; aI[nt] = zero8; }
  for (int kt = 0; kt < 16; ++kt) {
    const int k0 = kt * 4 + kb;
    v2f xr, xi, xin;
    xr.x = Xr[mrow * ES + k0];
    xr.y = Xr[mrow * ES + k0 + 1];
    xi.x = Xi[mrow * ES + k0];
    xi.y = Xi[mrow * ES + k0 + 1];
    xin.x = -xi.x;
    xin.y = -xi.y;
#pragma unroll
    for (int nt = 0; nt < 4; ++nt) {
      const int nc = nt * 16 + mlo;
      v2f yr, yi;
      yr.x = Yr[k0 * ES + nc];
      yr.y = Yr[(k0 + 1) * ES + nc];
      yi.x = Yi[k0 * ES + nc];
      yi.y = Yi[(k0 + 1) * ES + nc];
      aR[nt] = WMMA4(xr, yr, aR[nt]);    // + Xr*Yr
      aR[nt] = WMMA4(xin, yi, aR[nt]);   // - Xi*Yi
      aI[nt] = WMMA4(xr, yi, aI[nt]);    // + Xr*Yi
      aI[nt] = WMMA4(xi, yr, aI[nt]);    // + Xi*Yr
    }
  }
}

__launch_bounds__(128)
__global__ void expm_kernel(const float* __restrict__ theta,
                            float* __restrict__ psiR, float* __restrict__ psiI) {
  __shared__ float Ar[64 * ES], Ai[64 * ES], Er[64 * ES], Ei[64 * ES];
  __shared__ float rowsum[64];
  __shared__ int sSteps;
  const int tid  = threadIdx.x;
  const int b    = blockIdx.x;
  const int lane = tid & 31;
  const int wave = tid >> 5;

  // Phase 1: stage theta and per-generator metadata in LDS (reuse Ar/Ai).
  int* meta = (int*)Ai;
  for (int m = tid; m < 4095; m += 128) Ar[m] = theta[b * 4095 + m];
  for (int m = tid; m < 4095; m += 128) {
    const int word = m + 1;  // skip identity
    int xm = 0, yz = 0, ny = 0;
#pragma unroll
    for (int p = 0; p < 6; ++p) {
      const int d = (word >> (2 * p)) & 3;  // I=0,X=1,Y=2,Z=3 at bit position p
      xm |= ((d == 1) | (d == 2)) << p;
      yz |= (d >= 2) << p;
      ny += (d == 2);
    }
    meta[m] = xm | (yz << 6) | ((ny & 3) << 12);
  }
  for (int e = tid; e < 64 * ES; e += 128) { Er[e] = 0.0f; Ei[e] = 0.0f; }
  __syncthreads();

  // Phase 2: H[i][i^xm] += theta * i^(3*nY) * (-1)^popc(i&yz). One row/thread.
  if (tid < 64) {
    const int i = tid;
    float* hr = &Er[i * ES];
    float* hi = &Ei[i * ES];
    for (int m = 0; m < 4095; ++m) {
      float t = Ar[m];
      const int md = meta[m];
      const int xm = md & 63, yz = (md >> 6) & 63, c = (md >> 12) & 3;
      if (__popc(i & yz) & 1) t = -t;
      const int j = i ^ xm;
      switch (c) {
        case 0: hr[j] += t; break;   // *1
        case 1: hi[j] -= t; break;   // *(-i)
        case 2: hr[j] -= t; break;   // *(-1)
        default: hi[j] += t; break;  // *(+i)
      }
    }
    float rs = 0.0f;
    for (int j = 0; j < 64; ++j) rs += fabsf(hr[j]) + fabsf(hi[j]);
    rowsum[i] = rs;
  }
  __syncthreads();
  if (tid == 0) {
    float nrm = 0.0f;
    for (int j = 0; j < 64; ++j) nrm = fmaxf(nrm, rowsum[j]);
    int s = 0;
    while (nrm > 0.25f && s < 20) { nrm *= 0.5f; ++s; }
    sSteps = s;
  }
  __syncthreads();
  const int s = sSteps;
  const float sc = exp2f(-(float)s);

  // Phase 3: A = i*H/2^s (Ar/Ai free now), then E = I + A/9.
  for (int e = tid; e < 64 * ES; e += 128) {
    const float hrv = Er[e], hiv = Ei[e];
    Ar[e] = -hiv * sc;
    Ai[e] = hrv * sc;
  }
  __syncthreads();
  for (int e = tid; e < 64 * ES; e += 128) {
    Er[e] = Ar[e] * (1.0f / 9.0f);
    Ei[e] = Ai[e] * (1.0f / 9.0f);
  }
  __syncthreads();
  if (tid < 64) Er[tid * ES + tid] += 1.0f;
  __syncthreads();

  // Phase 4: Horner (E <- I + A*E/k, k=8..1) then s squarings (E <- E*E).
  const int mlo = lane & 15;
  const int kb = (lane < 16) ? 0 : 2;
  const int mrow = wave * 16 + mlo;
  const int total = 8 + s;
  for (int it = 0; it < total; ++it) {
    const bool horner = (it < 8);
    const float* Xr = horner ? (const float*)Ar : (const float*)Er;
    const float* Xi = horner ? (const float*)Ai : (const float*)Ei;
    v8f aR[4], aI[4];
    cmm_accum(Xr, Xi, Er, Ei, mrow, kb, mlo, aR, aI);
    __syncthreads();
    const float invk = horner ? 1.0f / (float)(8 - it) : 1.0f;
#pragma unroll
    for (int nt = 0; nt < 4; ++nt) {
      const int nc = nt * 16 + mlo;
#pragma unroll
      for (int v = 0; v < 8; ++v) {
        const int mr = wave * 16 + v + ((lane < 16) ? 0 : 8);
        float er = aR[nt][v] * invk;
        if (horner && mr == nc) er += 1.0f;
        Er[mr * ES + nc] = er;
        Ei[mr * ES + nc] = aI[nt][v] * invk;
      }
    }
    __syncthreads();
  }

  if (tid < 64) {
    psiR[b * 64 + tid] = Er[tid * ES];  // first column = exp(iH) e0
    psiI[b * 64 + tid] = Ei[tid * ES];
  }
}

// ---------------------------------------------------------------------------
// Two-qubit expectation values: q[b,w] = sum_r v_r^H H_w v_r, 15 wire pairs.
// ---------------------------------------------------------------------------
__launch_bounds__(64)
__global__ void expect_kernel(const float* __restrict__ psiR, const float* __restrict__ psiI,
                              const float* __restrict__ Aoff, const float* __restrict__ Boff,
                              const float* __restrict__ Ddiag, float* __restrict__ q) {
  __shared__ float pr[64], pi[64];
  const int b = blockIdx.x, tid = threadIdx.x;
  pr[tid] = psiR[b * 64 + tid];
  pi[tid] = psiI[b * 64 + tid];
  __syncthreads();
  if (tid < 15) {
    const int w = tid;
    int wa = 0, wb = 0, cnt = 0;
    for (int a0 = 0; a0 < 6; ++a0)
      for (int b0 = a0 + 1; b0 < 6; ++b0) {
        if (cnt == w) { wa = a0; wb = b0; }
        ++cnt;
      }
    const int pa = 5 - wa, pb = 5 - wb;
    int pos[4], np = 0;
    for (int p = 5; p >= 0; --p)
      if (p != pa && p != pb) pos[np++] = p;

    float ao[6], bo[6];
#pragma unroll
    for (int c = 0; c < 6; ++c) { ao[c] = Aoff[w * 6 + c]; bo[c] = Boff[w * 6 + c]; }
    const float d0 = 2.0f * Ddiag[w * 4 + 1];
    const float d1 = 2.0f * Ddiag[w * 4 + 2];
    const float d2 = 2.0f * Ddiag[w * 4 + 3];  // H[3][3] = 0
    const int ri[6] = {1, 2, 2, 3, 3, 3};
    const int ci[6] = {0, 0, 1, 0, 1, 2};

    float acc = 0.0f;
    for (int r = 0; r < 16; ++r) {
      int base = 0;
#pragma unroll
      for (int jj = 0; jj < 4; ++jj) base |= ((r >> (3 - jj)) & 1) << pos[jj];
      float vr[4], vi[4];
#pragma unroll
      for (int k = 0; k < 4; ++k) {
        const int idx = base | (((k >> 1) & 1) << pa) | ((k & 1) << pb);
        vr[k] = pr[idx];
        vi[k] = pi[idx];
      }
      acc += d0 * (vr[0] * vr[0] + vi[0] * vi[0]) +
             d1 * (vr[1] * vr[1] + vi[1] * vi[1]) +
             d2 * (vr[2] * vr[2] + vi[2] * vi[2]);
#pragma unroll
      for (int c = 0; c < 6; ++c) {
        const int i2 = ri[c], j2 = ci[c];
        const float x = vr[i2] * vr[j2] + vi[i2] * vi[j2];   // Re(conj(v_i) v_j)
        const float y = vr[i2] * vi[j2] - vi[i2] * vr[j2];   // Im(conj(v_i) v_j)
        acc += 2.0f * (ao[c] * x - bo[c] * y);
      }
    }
    q[b * 15 + w] = acc;
  }
}

// ---------------------------------------------------------------------------
extern "C" void kernel_launch(void* const* d_in, const int* in_sizes, int n_in,
                              void* d_out, int out_size, void* d_ws, size_t ws_size,
                              hipStream_t stream) {
  (void)in_sizes; (void)n_in; (void)out_size; (void)ws_size;
  const float* x     = (const float*)d_in[0];
  const float* W1    = (const float*)d_in[1];
  const float* b1    = (const float*)d_in[2];
  const float* W2    = (const float*)d_in[3];
  const float* b2    = (const float*)d_in[4];
  const float* Aoff  = (const float*)d_in[5];
  const float* Boff  = (const float*)d_in[6];
  const float* Ddiag = (const float*)d_in[7];
  const float* Wv1   = (const float*)d_in[8];
  const float* bv1   = (const float*)d_in[9];
  const float* Wv2   = (const float*)d_in[10];
  const float* bv2   = (const float*)d_in[11];
  // d_in[12] = pauli: unused (structure generated analytically).

  float* ws    = (float*)d_ws;
  float* h1    = ws;               // 512*256
  float* theta = h1 + 512 * 256;   // 512*4095
  float* psiR  = theta + 512 * 4095;
  float* psiI  = psiR + 512 * 64;
  float* qbuf  = psiI + 512 * 64;  // 512*15
  float* h2    = qbuf + 512 * 15;  // 512*256
  float* out   = (float*)d_out;    // 512*512

  gemm_f32_kernel<<<dim3(8, 4), 128, 0, stream>>>(x, W1, b1, h1, 512, 256, 768, 1);
  gemm_f32_kernel<<<dim3(8, 64), 128, 0, stream>>>(h1, W2, b2, theta, 512, 4095, 256, 0);
  expm_kernel<<<dim3(512), 128, 0, stream>>>(theta, psiR, psiI);
  expect_kernel<<<dim3(512), 64, 0, stream>>>(psiR, psiI, Aoff, Boff, Ddiag, qbuf);
  gemm_f32_kernel<<<dim3(8, 4), 128, 0, stream>>>(qbuf, Wv1, bv1, h2, 512, 256, 15, 1);
  gemm_f32_kernel<<<dim3(8, 8), 128, 0, stream>>>(h2, Wv2, bv2, out, 512, 512, 256, 0);
}